// GraphNeuralODE_12206297055732
// MI455X (gfx1250) — compile-verified
//
#include <hip/hip_runtime.h>
#include <math.h>

#define N_NODES 20000
#define E_EDGES 320000
#define HDIM    64
#define TPTS    8
#define NH      (N_NODES * HDIM)

// gemm tiling: 1250 row tiles (16 nodes each) = 250 groups of 5; 4 col tiles.
// 1000 waves -> 125 blocks * 8 waves.
#define ROWTILES_PER_WAVE 5
#define GEMM_BLOCKS 125

typedef __attribute__((ext_vector_type(2))) float v2f;
typedef __attribute__((ext_vector_type(8))) float v8f;

// ---------------------------------------------------------------------------
// y0[n,h] = x0[n] * W_init[h] + b_init[h]
// ---------------------------------------------------------------------------
__global__ void encode_kernel(const float* __restrict__ x0,
                              const float* __restrict__ Wi,
                              const float* __restrict__ bi,
                              float* __restrict__ y) {
    int tid = blockIdx.x * blockDim.x + threadIdx.x;
    if (tid < NH) {
        int n = tid >> 6;
        int h = tid & 63;
        y[tid] = x0[n] * Wi[h] + bi[h];
    }
}

// ---------------------------------------------------------------------------
// agg = 0 (run once at call start; thereafter the gemm kernel re-zeroes agg
// in its A-load pass, so every call starts deterministic with no leftover
// state dependence).
// ---------------------------------------------------------------------------
__global__ void zero_kernel(float4* __restrict__ p) {
    int i = blockIdx.x * 256 + threadIdx.x;
    p[i] = make_float4(0.f, 0.f, 0.f, 0.f);
}

// ---------------------------------------------------------------------------
// Edge gather+scale+scatter: 16 lanes per edge, float4 row segment each.
// w_e = c0[e] + (c1[e]-c0[e])*frac ;  agg[dst[e]] += y[src[e]] * w_e
// L2-atomic bound; y (5.12MB) and agg (5.12MB) are L2-resident on MI455X.
// Exact grid: E*16/256 blocks.
// ---------------------------------------------------------------------------
__global__ void edge_scatter_kernel(const int* __restrict__ src,
                                    const int* __restrict__ dst,
                                    const float* __restrict__ c0,
                                    const float* __restrict__ c1,
                                    float frac,
                                    const float* __restrict__ y,
                                    float* __restrict__ agg) {
    int tid = blockIdx.x * 256 + threadIdx.x;
    int e = tid >> 4;            // edge index
    int q = (tid & 15) << 2;     // column offset (0,4,...,60)
    float a = c0[e];
    float w = a + (c1[e] - a) * frac;
    int s = src[e];
    int d = dst[e];
    const float4 v = *reinterpret_cast<const float4*>(y + (size_t)s * HDIM + q);
    float* ap = agg + (size_t)d * HDIM + q;
    atomicAdd(ap + 0, v.x * w);
    atomicAdd(ap + 1, v.y * w);
    atomicAdd(ap + 2, v.z * w);
    atomicAdd(ap + 3, v.w * w);
}

// ---------------------------------------------------------------------------
// Fused vf + RK4 stage update, fp32 WMMA 16x16x4:
//   k       = tanh( agg @ W_msg + yin @ W_self + b_vf )      (in-register)
//   accout  = accin + wk * k
//   ytmpout = ybase + ss * k            (if write_tmp)
//   agg     = 0                         (rewritten during A-load pass)
//
// One wave owns one 16-col tile and ROWTILES_PER_WAVE row tiles; both B
// fragments (W_msg, W_self K-slices) are register-cached across row tiles
// (16 x v2f each), cutting weight re-fetch 5x vs tile-per-wave.
//
// Fragment layouts (wave32):
//   A (16x4):  lane m=lane&15 -> row; lane half selects K pair (float2 load)
//   B (4x16):  lane n=lane&15 -> col; same K striping across lane halves
//   C (16x16): VGPR r -> row r (lanes 0-15) / r+8 (lanes 16-31), col=lane&15
// Exact tiling -> EXEC all-ones for every WMMA wave.
// ---------------------------------------------------------------------------
__global__ void __launch_bounds__(256)
gemm_rk4_kernel(float* __restrict__ agg,
                const float* __restrict__ yin,
                const float* __restrict__ ybase,
                const float* __restrict__ accin,
                float* __restrict__ accout,
                float* __restrict__ ytmpout,
                const float* __restrict__ Wmsg,
                const float* __restrict__ Wself,
                const float* __restrict__ bvf,
                float wk, float ss, int write_tmp) {
    int lane = threadIdx.x & 31;
    int g    = blockIdx.x * 8 + (threadIdx.x >> 5);   // 0..999
    int ct   = g & 3;                                 // col tile 0..3
    int rg   = g >> 2;                                // row group 0..249
    int col_base = ct * 16;
    int m  = lane & 15;
    int kh = (lane >> 4) << 1;                        // 0 or 2
    int rowoff = (lane >> 4) << 3;                    // 0 or 8

    // Register-cached B fragments for K=64 of both weight matrices.
    v2f bm[16], bs[16];
    #pragma unroll
    for (int i = 0; i < 16; ++i) {
        int k = i * 4 + kh;
        bm[i].x = Wmsg [(size_t)(k + 0) * HDIM + col_base + m];
        bm[i].y = Wmsg [(size_t)(k + 1) * HDIM + col_base + m];
        bs[i].x = Wself[(size_t)(k + 0) * HDIM + col_base + m];
        bs[i].y = Wself[(size_t)(k + 1) * HDIM + col_base + m];
    }
    float bias = bvf[col_base + m];

    v2f zv; zv.x = 0.f; zv.y = 0.f;

    for (int rr = 0; rr < ROWTILES_PER_WAVE; ++rr) {
        int row_base = (rg * ROWTILES_PER_WAVE + rr) * 16;
        float*       arow = agg + (size_t)(row_base + m) * HDIM;
        const float* yrow = yin + (size_t)(row_base + m) * HDIM;

        v8f c = {};
        #pragma unroll
        for (int i = 0; i < 16; ++i) {
            v2f af = *reinterpret_cast<const v2f*>(arow + i * 4 + kh);
            c = __builtin_amdgcn_wmma_f32_16x16x4_f32(false, af, false, bm[i],
                                                      (short)0, c, false, false);
        }
        // Re-zero the agg elements this lane just consumed (exact coverage:
        // each (row,k) element is read by exactly one lane exactly once).
        #pragma unroll
        for (int i = 0; i < 16; ++i) {
            *reinterpret_cast<v2f*>(arow + i * 4 + kh) = zv;
        }
        #pragma unroll
        for (int i = 0; i < 16; ++i) {
            v2f af = *reinterpret_cast<const v2f*>(yrow + i * 4 + kh);
            c = __builtin_amdgcn_wmma_f32_16x16x4_f32(false, af, false, bs[i],
                                                      (short)0, c, false, false);
        }

        // Fused RK4 epilogue (no kbuf round-trip).
        #pragma unroll
        for (int r = 0; r < 8; ++r) {
            size_t idx = (size_t)(row_base + rowoff + r) * HDIM + col_base + m;
            float kv = tanhf(c[r] + bias);
            accout[idx] = accin[idx] + wk * kv;
            if (write_tmp) ytmpout[idx] = ybase[idx] + ss * kv;
        }
    }
}

// ---------------------------------------------------------------------------
// out[n] = dot(y[n,:], W_fin) + b_fin
// ---------------------------------------------------------------------------
__global__ void decode_kernel(const float* __restrict__ y,
                              const float* __restrict__ Wfin,
                              const float* __restrict__ bfin,
                              float* __restrict__ out) {
    int n = blockIdx.x * 256 + threadIdx.x;
    if (n >= N_NODES) return;
    const float4* row = reinterpret_cast<const float4*>(y + (size_t)n * HDIM);
    const float4* wf  = reinterpret_cast<const float4*>(Wfin);
    float s = 0.f;
    #pragma unroll
    for (int i = 0; i < HDIM / 4; ++i) {
        float4 a = row[i];
        float4 b = wf[i];
        s += a.x * b.x + a.y * b.y + a.z * b.z + a.w * b.w;
    }
    out[n] = s + bfin[0];
}

// ---------------------------------------------------------------------------
extern "C" void kernel_launch(void* const* d_in, const int* in_sizes, int n_in,
                              void* d_out, int out_size, void* d_ws, size_t ws_size,
                              hipStream_t stream) {
    (void)in_sizes; (void)n_in; (void)out_size; (void)ws_size;

    const float* ts     = (const float*)d_in[0]; (void)ts;  // arange(8)
    const float* coeffs = (const float*)d_in[1];            // [T,E]
    const float* x0     = (const float*)d_in[2];            // [N,1]
    const int*   adj    = (const int*)  d_in[3];            // [2,E]
    const float* W_init = (const float*)d_in[4];            // [H,1]
    const float* b_init = (const float*)d_in[5];            // [H]
    const float* W_msg  = (const float*)d_in[6];            // [H,H]
    const float* W_self = (const float*)d_in[7];            // [H,H]
    const float* b_vf   = (const float*)d_in[8];            // [H]
    const float* W_fin  = (const float*)d_in[9];            // [1,H]
    const float* b_fin  = (const float*)d_in[10];           // [1]
    float* out = (float*)d_out;

    const int* src = adj;
    const int* dst = adj + E_EDGES;

    float* ws  = (float*)d_ws;
    float* y   = ws + 0 * (size_t)NH;
    float* acc = ws + 1 * (size_t)NH;
    float* tA  = ws + 2 * (size_t)NH;   // ytmp double-buffer A
    float* tB  = ws + 3 * (size_t)NH;   // ytmp double-buffer B
    float* agg = ws + 4 * (size_t)NH;

    encode_kernel<<<(NH + 255) / 256, 256, 0, stream>>>(x0, W_init, b_init, y);
    zero_kernel<<<NH / 4 / 256, 256, 0, stream>>>((float4*)agg);
    decode_kernel<<<(N_NODES + 255) / 256, 256, 0, stream>>>(y, W_fin, b_fin, out);

    const float dt = 1.0f;   // ts = arange(T)

    for (int a = 0; a < TPTS - 1; ++a) {
        int   i_end = (a + 1 <= TPTS - 2) ? (a + 1) : (TPTS - 2);
        float f_end = (float)((a + 1) - i_end);

        struct Stage {
            int i; float frac; float wk; float ss;
            const float* yin; const float* accin; float* tmpout; int wt;
        };
        const Stage st[4] = {
            { a,     0.0f,  dt / 6.0f, 0.5f * dt, y,  y,   tA, 1 },
            { a,     0.5f,  dt / 3.0f, 0.5f * dt, tA, acc, tB, 1 },
            { a,     0.5f,  dt / 3.0f, 1.0f * dt, tB, acc, tA, 1 },
            { i_end, f_end, dt / 6.0f, 0.0f,      tA, acc, tB, 0 },
        };

        for (int s = 0; s < 4; ++s) {
            edge_scatter_kernel<<<(E_EDGES * 16) / 256, 256, 0, stream>>>(
                src, dst,
                coeffs + (size_t)st[s].i * E_EDGES,
                coeffs + (size_t)(st[s].i + 1) * E_EDGES,
                st[s].frac, st[s].yin, agg);

            gemm_rk4_kernel<<<GEMM_BLOCKS, 256, 0, stream>>>(
                agg, st[s].yin, y, st[s].accin, acc, st[s].tmpout,
                W_msg, W_self, b_vf,
                st[s].wk, st[s].ss, st[s].wt);
        }

        decode_kernel<<<(N_NODES + 255) / 256, 256, 0, stream>>>(
            acc, W_fin, b_fin, out + (size_t)(a + 1) * N_NODES);

        float* tswap = y; y = acc; acc = tswap;
    }
}